// SparseConvUnet_9569187135706
// MI455X (gfx1250) — compile-verified
//
#include <hip/hip_runtime.h>
#include <hip/hip_bf16.h>
#include <stdint.h>

#define MAX_ATOMS 14
#define CHN 128                 // channels per row -> 512 bytes per row
#define WAVES_PER_BLOCK 8       // 256 threads, wave32
#define BATCH 8                 // rows per wave (depth-8 async pipeline)

#define AS1 __attribute__((address_space(1)))
#define AS3 __attribute__((address_space(3)))

typedef int v4i __attribute__((vector_size(16)));   // matches builtin param type

#if defined(__has_builtin)
#if __has_builtin(__builtin_amdgcn_global_load_async_to_lds_b128) && \
    __has_builtin(__builtin_amdgcn_global_store_async_from_lds_b128)
#define HAVE_ASYNC_LDS 1
#endif
#endif
#ifndef HAVE_ASYNC_LDS
#define HAVE_ASYNC_LDS 0
#endif

__device__ __forceinline__ void async_copy_g2l_b128(void* gsrc, void* lds) {
#if HAVE_ASYNC_LDS
  __builtin_amdgcn_global_load_async_to_lds_b128(
      (AS1 v4i*)(v4i*)gsrc, (AS3 v4i*)(v4i*)lds, 0, 0);
#else
  *(float4*)lds = *(const float4*)gsrc;
#endif
}

__device__ __forceinline__ void async_copy_l2g_b128(void* gdst, void* lds) {
#if HAVE_ASYNC_LDS
  __builtin_amdgcn_global_store_async_from_lds_b128(
      (AS1 v4i*)(v4i*)gdst, (AS3 v4i*)(v4i*)lds, 0, 0);
#else
  *(float4*)gdst = *(const float4*)lds;
#endif
}

__device__ __forceinline__ void wait_async_zero() {
#if defined(__has_builtin) && __has_builtin(__builtin_amdgcn_s_wait_asynccnt)
  __builtin_amdgcn_s_wait_asynccnt(0);
#else
  asm volatile("s_wait_asynccnt 0" ::: "memory");
#endif
}

// ---------------------------------------------------------------------------
// Kernel: ragged gather -> padded [N,14,128] via the async LDS engine.
// One wave per 8 output rows; each row = 512 B = 32 lanes x b128.
// Mask (s < atom_num[r]) is wave-uniform -> no divergence.
// All row indexing is 32-bit (totalRows = 1.4M, element offsets <= 179.2M);
// (r, s) advance incrementally so the unrolled loop has no division at all.
// ---------------------------------------------------------------------------
__global__ __launch_bounds__(256)
void SCU_gather_kernel(const float* __restrict__ feature,
                       const int* __restrict__ atom_num,
                       const int* __restrict__ offsets,
                       float* __restrict__ out,
                       int n_res, int n_feat_rows) {
  __shared__ float4 stage[WAVES_PER_BLOCK * BATCH * 32]; // 32 KB
  const int lane = (int)(threadIdx.x & 31u);
  const int wave = (int)(threadIdx.x >> 5);
  const int totalRows = n_res * MAX_ATOMS;
  const int row0 = ((int)blockIdx.x * WAVES_PER_BLOCK + wave) * BATCH;
  if (row0 >= totalRows) return;

  float4* bufs = &stage[(wave * BATCH) * 32];

  // (r, s) for row0; single 32-bit const-division, then incremental updates.
  int r = (int)((unsigned)row0 / MAX_ATOMS);
  int s = row0 - r * MAX_ATOMS;

  // Phase 1: fire up to BATCH async b128 gathers into this wave's LDS staging.
  int maskbits = 0;
  const int maxRow = n_feat_rows - 1;
#pragma unroll
  for (int d = 0; d < BATCH; ++d) {
    const int row = row0 + d;
    if (row < totalRows) {
      const int cnt = atom_num[r];
      if (s < cnt) {
        int src_row = offsets[r] + s;
        src_row = (src_row > maxRow) ? maxRow : src_row;   // match reference clip
        src_row = (src_row < 0) ? 0 : src_row;
        const float4* src = (const float4*)(feature + (size_t)src_row * CHN);
        async_copy_g2l_b128((void*)&src[lane], (void*)&bufs[d * 32 + lane]);
        maskbits |= (1 << d);
      }
    }
    if (++s == MAX_ATOMS) { s = 0; ++r; }
  }

  wait_async_zero();   // all gathered rows landed in LDS

  // Phase 2: async-store real rows from LDS; plain b128 zero-stores otherwise.
  float4 z; z.x = 0.f; z.y = 0.f; z.z = 0.f; z.w = 0.f;
#pragma unroll
  for (int d = 0; d < BATCH; ++d) {
    const int row = row0 + d;
    if (row >= totalRows) break;
    float4* dst = (float4*)(out + (size_t)row * CHN);
    if (maskbits & (1 << d)) {
      async_copy_l2g_b128((void*)&dst[lane], (void*)&bufs[d * 32 + lane]);
    } else {
      dst[lane] = z;
    }
  }
  // S_ENDPGM performs an implicit wait-idle (drains ASYNCcnt/STOREcnt).
}

// ---------------------------------------------------------------------------
// Kernel: exclusive prefix sum of atom_num -> offsets (single block, wave32
// shuffle scans; ~98 chunks of 1024, negligible vs. the 1.43 GB copy).
// ---------------------------------------------------------------------------
__global__ __launch_bounds__(1024)
void SCU_scan_kernel(const int* __restrict__ counts, int* __restrict__ offsets, int n) {
  __shared__ int warpTot[32];
  __shared__ int warpPre[32];
  __shared__ int carry;
  __shared__ int chunkTot;
  const int tid  = threadIdx.x;
  const int lane = tid & 31;
  const int wid  = tid >> 5;
  if (tid == 0) carry = 0;
  __syncthreads();
  for (int base = 0; base < n; base += 1024) {
    const int i = base + tid;
    const int v = (i < n) ? counts[i] : 0;
    int x = v;
#pragma unroll
    for (int d = 1; d < 32; d <<= 1) {
      int t = __shfl_up(x, d, 32);
      if (lane >= d) x += t;
    }
    if (lane == 31) warpTot[wid] = x;
    __syncthreads();
    if (wid == 0) {
      int w = warpTot[lane];
      int y = w;
#pragma unroll
      for (int d = 1; d < 32; d <<= 1) {
        int t = __shfl_up(y, d, 32);
        if (lane >= d) y += t;
      }
      warpPre[lane] = y - w;
      if (lane == 31) chunkTot = y;
    }
    __syncthreads();
    if (i < n) offsets[i] = (x - v) + warpPre[wid] + carry;
    __syncthreads();
    if (tid == 0) carry += chunkTot;
    __syncthreads();
  }
}

// ---------------------------------------------------------------------------
// Kernel: pass-through of label_binary (int32 bits appended after floats).
// ---------------------------------------------------------------------------
__global__ __launch_bounds__(256)
void SCU_label_kernel(const int* __restrict__ label, int* __restrict__ out, int n) {
  const int i = blockIdx.x * blockDim.x + threadIdx.x;
  if (i < n) out[i] = label[i];
}

extern "C" void kernel_launch(void* const* d_in, const int* in_sizes, int n_in,
                              void* d_out, int out_size, void* d_ws, size_t ws_size,
                              hipStream_t stream) {
  (void)n_in; (void)out_size; (void)ws_size;
  const float* feature  = (const float*)d_in[0];
  const int*   atom_num = (const int*)d_in[1];
  const int*   label    = (const int*)d_in[2];

  const int n_res       = in_sizes[1];
  const int n_feat_rows = in_sizes[0] / CHN;   // = n_res * MAX_ATOMS

  int*   offsets = (int*)d_ws;                 // n_res int32 scratch
  float* out     = (float*)d_out;

  // 1) offsets = exclusive cumsum(atom_num)
  SCU_scan_kernel<<<1, 1024, 0, stream>>>(atom_num, offsets, n_res);

  // 2) gather-to-padded copy (HBM-bound: ~1.43 GB -> ~62 us at 23.3 TB/s)
  const int totalRows = n_res * MAX_ATOMS;
  const int rowsPerBlock = WAVES_PER_BLOCK * BATCH;
  const int grid = (totalRows + rowsPerBlock - 1) / rowsPerBlock;
  SCU_gather_kernel<<<grid, 256, 0, stream>>>(feature, atom_num, offsets, out,
                                              n_res, n_feat_rows);

  // 3) labels appended (bit-copy of int32 into the tail of d_out)
  int* label_out = (int*)(out + (size_t)totalRows * CHN);
  SCU_label_kernel<<<(n_res + 255) / 256, 256, 0, stream>>>(label, label_out, n_res);
}